// GNN_3186865733926
// MI455X (gfx1250) — compile-verified
//
#include <hip/hip_runtime.h>
#include <cstdint>

// ---------------------------------------------------------------------------
// Hetero-GAT (2 layers) for MI455X / gfx1250, wave32, WMMA f32 16x16x4.
//
// Assumed d_in layout (setup_inputs dict order, params pytree flattened in
// JAX sorted-key order):
//   0: x_user [NU,32] f32   1: x_review [NR,256] f32   2: x_product [NP,64] f32
//   3..22 : conv1 params, edge order (rev_by, reviews, writes, written_by),
//           each (Wd[128,128], Ws[128,128], ad[128], as[128], b[128])
//   23..42: conv2 params, same layout
//   43..48: lin params (product W[64,128] b, review W[256,128] b, user W[32,128] b)
//   49..56: writes_src, writes_dst, reviews_src, reviews_dst,
//           rev_by_src, rev_by_dst, written_by_src, written_by_dst   (int32)
// Output: concat(user[NU,128], review[NR,128], product[NP,128]) f32.
// ---------------------------------------------------------------------------

#define H 128
#define NEG_SLOPE 0.2f

typedef __attribute__((ext_vector_type(2))) float v2f;
typedef __attribute__((ext_vector_type(8))) float v8f;

// ---------------- WMMA GEMM:  C[M,128] = A[M,K] @ W[K,128] (+ bias) --------
// One wave per 16x16 output tile; 8 waves/block cover all 128 columns.
// Exact fp32 via V_WMMA_F32_16X16X4_F32 (A: 16x4 f32 = 2 VGPR/lane).
// Tail handling: A row index is CLAMPED (not predicated) -- rows >= M only
// feed output rows that are never stored, so no exec masking is needed and
// all A loads stay unconditional/clauseable.
__global__ void __launch_bounds__(256)
gemm128_wmma(const float* __restrict__ A, const float* __restrict__ W,
             const float* __restrict__ bias, float* __restrict__ C,
             int M, int K) {
    const int wave = threadIdx.x >> 5;      // 0..7 -> 16-col tile
    const int lane = threadIdx.x & 31;
    const int half = lane >> 4;             // 0/1 (K-pair select per ISA layout)
    const int l16  = lane & 15;
    const int rowBase = blockIdx.x * 16;
    const int colBase = wave * 16;

    int mRow = rowBase + l16;
    if (mRow > M - 1) mRow = M - 1;         // clamp: safe, rows >=M not stored
    const float* arow = A + (size_t)mRow * K + half * 2;
    const float* wcol = W + (size_t)(half * 2) * H + colBase + l16;

    v8f acc = {};
    for (int k0 = 0; k0 < K; k0 += 4) {
        v2f a;
        a.x = arow[k0];
        a.y = arow[k0 + 1];
        v2f b;
        const float* wp = wcol + (size_t)k0 * H;
        b.x = wp[0];
        b.y = wp[H];
        // (neg_a, A, neg_b, B, c_mod, C, reuse_a, reuse_b)
        acc = __builtin_amdgcn_wmma_f32_16x16x4_f32(false, a, false, b,
                                                    (short)0, acc, false, false);
    }

    const int c = colBase + l16;
    const float bv = bias ? bias[c] : 0.f;
#pragma unroll
    for (int r = 0; r < 8; ++r) {
        const int m = rowBase + r + half * 8;   // C layout: VGPR r -> M=r / r+8
        if (m < M) C[(size_t)m * H + c] = acc[r] + bv;
    }
}

// ---------------- v[i] = sum_j W[i,j] * a[j]   (128x128 @ 128) -------------
__global__ void colvec_kernel(const float* __restrict__ W,
                              const float* __restrict__ a,
                              float* __restrict__ v) {
    const int i = threadIdx.x;               // 128 threads
    const float* wr = W + (size_t)i * H;
    float acc = 0.f;
    for (int j = 0; j < H; ++j) acc += wr[j] * a[j];
    v[i] = acc;
}

// ---------------- s[n] = X[n,:] . v   (wave per row) -----------------------
__global__ void __launch_bounds__(256)
rowdot_kernel(const float* __restrict__ X, int N, int K,
              const float* __restrict__ v, float* __restrict__ s) {
    const int row  = blockIdx.x * 8 + (threadIdx.x >> 5);
    const int lane = threadIdx.x & 31;
    if (row >= N) return;
    const float* xp = X + (size_t)row * K;
    float acc = 0.f;
    for (int k = lane; k < K; k += 32) acc += xp[k] * v[k];
#pragma unroll
    for (int off = 16; off; off >>= 1) acc += __shfl_xor(acc, off, 32);
    if (lane == 0) s[row] = acc;
}

// ---------------- order-preserving float<->uint for atomicMax --------------
__device__ __forceinline__ unsigned fenc(float f) {
    unsigned u = __float_as_uint(f);
    return (u & 0x80000000u) ? ~u : (u | 0x80000000u);
}
__device__ __forceinline__ float fdec(unsigned u) {
    return __uint_as_float((u & 0x80000000u) ? (u ^ 0x80000000u) : ~u);
}

// ---------------- edge logits + segment max over dst -----------------------
__global__ void __launch_bounds__(256)
logit_max_kernel(const int* __restrict__ src, const int* __restrict__ dst,
                 const float* __restrict__ sS, const float* __restrict__ sD,
                 float* __restrict__ elog, unsigned* __restrict__ mEnc, int E) {
    const int e = blockIdx.x * blockDim.x + threadIdx.x;
    if (e >= E) return;
    float x = sS[src[e]] + sD[dst[e]];
    x = (x > 0.f) ? x : NEG_SLOPE * x;       // leaky_relu
    elog[e] = x;
    atomicMax(&mEnc[dst[e]], fenc(x));
}

// ---------------- ex = exp(logit - m[dst]);  denom[dst] += ex --------------
__global__ void __launch_bounds__(256)
expsum_kernel(const int* __restrict__ dst, float* __restrict__ elog,
              const unsigned* __restrict__ mEnc, float* __restrict__ denom,
              int E) {
    const int e = blockIdx.x * blockDim.x + threadIdx.x;
    if (e >= E) return;
    const int d = dst[e];
    const float ex = expf(elog[e] - fdec(mEnc[d]));
    elog[e] = ex;                            // overwrite logit with ex
    unsafeAtomicAdd(&denom[d], ex);
}

// ---------------- out[dst] += (ex/denom[dst]) * HS[src]  (wave per edge) ---
__global__ void __launch_bounds__(256)
aggregate_kernel(const int* __restrict__ src, const int* __restrict__ dst,
                 const float* __restrict__ elog, const float* __restrict__ denom,
                 const float* __restrict__ HS, float* __restrict__ out, int E) {
    const int e = blockIdx.x * 8 + (threadIdx.x >> 5);
    if (e >= E) return;
    const int lane = threadIdx.x & 31;
    const int s = src[e], d = dst[e];
    const float coef = elog[e] / denom[d];
    const float* hp = HS + (size_t)s * H;
    float* op = out + (size_t)d * H;
#pragma unroll
    for (int j = 0; j < 4; ++j) {
        const int c = lane + 32 * j;
        unsafeAtomicAdd(&op[c], coef * hp[c]);
    }
}

// ---------------- out[n*128+c] = b1[c] (+ b2[c]) ---------------------------
__global__ void __launch_bounds__(256)
bias_init_kernel(float* __restrict__ out, const float* __restrict__ b1,
                 const float* __restrict__ b2, size_t n) {
    const size_t i = (size_t)blockIdx.x * blockDim.x + threadIdx.x;
    if (i >= n) return;
    const int c = (int)(i & (H - 1));
    float v = b1[c];
    if (b2) v += b2[c];
    out[i] = v;
}

__global__ void __launch_bounds__(256)
relu_kernel(float* __restrict__ x, size_t n) {
    const size_t i = (size_t)blockIdx.x * blockDim.x + threadIdx.x;
    if (i >= n) return;
    const float v = x[i];
    x[i] = v > 0.f ? v : 0.f;
}

// ===========================================================================
extern "C" void kernel_launch(void* const* d_in, const int* in_sizes, int n_in,
                              void* d_out, int out_size, void* d_ws, size_t ws_size,
                              hipStream_t stream) {
    (void)n_in; (void)out_size; (void)ws_size;

    const int DU = 32, DR = 256, DP = 64;
    const int NU = in_sizes[0] / DU;
    const int NR = in_sizes[1] / DR;
    const int NP = in_sizes[2] / DP;

    const float* X0[3] = { (const float*)d_in[0], (const float*)d_in[1],
                           (const float*)d_in[2] };
    const int    D0[3] = { DU, DR, DP };
    const int    Nn[3] = { NU, NR, NP };     // 0=user, 1=review, 2=product
    enum { T_USER = 0, T_REVIEW = 1, T_PRODUCT = 2 };

    auto F = [&](int i) { return (const float*)d_in[i]; };
    auto I = [&](int i) { return (const int*)d_in[i]; };

    // params: conv c (0/1), edge flatten-index fe; per-gat (Wd,Ws,ad,as,b)
    const int P0 = 3, LIN0 = 43, EIDX = 49;
    auto pWd = [&](int c, int fe) { return F(P0 + c * 20 + fe * 5 + 0); };
    auto pWs = [&](int c, int fe) { return F(P0 + c * 20 + fe * 5 + 1); };
    auto pAd = [&](int c, int fe) { return F(P0 + c * 20 + fe * 5 + 2); };
    auto pAs = [&](int c, int fe) { return F(P0 + c * 20 + fe * 5 + 3); };
    auto pB  = [&](int c, int fe) { return F(P0 + c * 20 + fe * 5 + 4); };

    // edge types: {flatten-index, srcType, dstType, src-array-idx, dst-array-idx}
    struct ET { int fe, srcT, dstT, si, di; };
    const ET WRITES     = { 2, T_USER,    T_REVIEW,  EIDX + 0, EIDX + 1 };
    const ET REVIEWS    = { 1, T_REVIEW,  T_PRODUCT, EIDX + 2, EIDX + 3 };
    const ET REV_BY     = { 0, T_PRODUCT, T_REVIEW,  EIDX + 4, EIDX + 5 };
    const ET WRITTEN_BY = { 3, T_REVIEW,  T_USER,    EIDX + 6, EIDX + 7 };
    const ET ets[4] = { WRITES, REV_BY, REVIEWS, WRITTEN_BY };

    int Emax = 0;
    for (int k = 0; k < 4; ++k)
        if (in_sizes[ets[k].si] > Emax) Emax = in_sizes[ets[k].si];

    // ---- workspace carve ----------------------------------------------------
    char* wsp = (char*)d_ws;
    auto carve = [&](size_t bytes) -> void* {
        void* p = wsp; wsp += (bytes + 255) & ~(size_t)255; return p;
    };
    float* bufA[3], * bufB[3];
    for (int t = 0; t < 3; ++t) bufA[t] = (float*)carve((size_t)Nn[t] * H * 4);
    for (int t = 0; t < 3; ++t) bufB[t] = (float*)carve((size_t)Nn[t] * H * 4);
    float*    HS    = (float*)carve((size_t)NR * H * 4);   // max Ns = NR
    float*    sS    = (float*)carve((size_t)NR * 4);
    float*    sD    = (float*)carve((size_t)NR * 4);
    unsigned* mEnc  = (unsigned*)carve((size_t)NR * 4);
    float*    denom = (float*)carve((size_t)NR * 4);
    float*    elog  = (float*)carve((size_t)Emax * 4);
    float*    wvS   = (float*)carve(H * 4);
    float*    wvD   = (float*)carve(H * 4);

    // ---- stage 0: linear projections into H=128 ----------------------------
    // lin flatten order: product(W,b), review(W,b), user(W,b)
    const float* linW[3] = { F(LIN0 + 4), F(LIN0 + 2), F(LIN0 + 0) }; // user,review,product
    const float* linB[3] = { F(LIN0 + 5), F(LIN0 + 3), F(LIN0 + 1) };
    for (int t = 0; t < 3; ++t)
        gemm128_wmma<<<(Nn[t] + 15) / 16, 256, 0, stream>>>(
            X0[t], linW[t], linB[t], bufA[t], Nn[t], D0[t]);

    // ---- one GAT edge-type pass --------------------------------------------
    auto run_gat = [&](const float* Xs, int Ns, const float* Xd, int Nd,
                       const int* src, const int* dst, int E,
                       const float* Ws, const float* Wd,
                       const float* as_, const float* ad_, float* Out) {
        colvec_kernel<<<1, H, 0, stream>>>(Ws, as_, wvS);   // Ws @ as
        colvec_kernel<<<1, H, 0, stream>>>(Wd, ad_, wvD);   // Wd @ ad
        rowdot_kernel<<<(Ns + 7) / 8, 256, 0, stream>>>(Xs, Ns, H, wvS, sS);
        rowdot_kernel<<<(Nd + 7) / 8, 256, 0, stream>>>(Xd, Nd, H, wvD, sD);
        gemm128_wmma<<<(Ns + 15) / 16, 256, 0, stream>>>(Xs, Ws, nullptr, HS, Ns, H);
        hipMemsetAsync(mEnc, 0, (size_t)Nd * 4, stream);    // fenc(-NaN) floor
        hipMemsetAsync(denom, 0, (size_t)Nd * 4, stream);
        logit_max_kernel<<<(E + 255) / 256, 256, 0, stream>>>(src, dst, sS, sD,
                                                              elog, mEnc, E);
        expsum_kernel<<<(E + 255) / 256, 256, 0, stream>>>(dst, elog, mEnc,
                                                           denom, E);
        aggregate_kernel<<<(E + 7) / 8, 256, 0, stream>>>(src, dst, elog, denom,
                                                          HS, Out, E);
    };

    // ---- one hetero conv layer ---------------------------------------------
    auto run_layer = [&](float* const Xin[3], float* const Xout[3], int c,
                         bool relu) {
        // init dst accumulators with summed GAT biases (HeteroConv aggr='sum')
        {
            size_t n = (size_t)NR * H;
            bias_init_kernel<<<(unsigned)((n + 255) / 256), 256, 0, stream>>>(
                Xout[T_REVIEW], pB(c, WRITES.fe), pB(c, REV_BY.fe), n);
            n = (size_t)NP * H;
            bias_init_kernel<<<(unsigned)((n + 255) / 256), 256, 0, stream>>>(
                Xout[T_PRODUCT], pB(c, REVIEWS.fe), nullptr, n);
            n = (size_t)NU * H;
            bias_init_kernel<<<(unsigned)((n + 255) / 256), 256, 0, stream>>>(
                Xout[T_USER], pB(c, WRITTEN_BY.fe), nullptr, n);
        }
        for (int k = 0; k < 4; ++k) {
            const ET& et = ets[k];
            run_gat(Xin[et.srcT], Nn[et.srcT], Xin[et.dstT], Nn[et.dstT],
                    I(et.si), I(et.di), in_sizes[et.si],
                    pWs(c, et.fe), pWd(c, et.fe), pAs(c, et.fe), pAd(c, et.fe),
                    Xout[et.dstT]);
        }
        if (relu)
            for (int t = 0; t < 3; ++t) {
                size_t n = (size_t)Nn[t] * H;
                relu_kernel<<<(unsigned)((n + 255) / 256), 256, 0, stream>>>(
                    Xout[t], n);
            }
    };

    // conv1: bufA -> bufB (+relu)
    run_layer(bufA, bufB, 0, true);

    // conv2: bufB -> d_out directly (output order: user, review, product)
    float* outp = (float*)d_out;
    float* Xout2[3] = { outp,                               // user
                        outp + (size_t)NU * H,              // review
                        outp + (size_t)(NU + NR) * H };     // product
    run_layer(bufB, Xout2, 1, false);
}